// TrafficPredictorGNN_58351425683571
// MI455X (gfx1250) — compile-verified
//
#include <hip/hip_runtime.h>
#include <math.h>

typedef __attribute__((ext_vector_type(2))) float v2f;
typedef __attribute__((ext_vector_type(8))) float v8f;

// ---------------- edge / node prep kernels (bandwidth-bound part) ----------------

__global__ void k_init_deg(float* deg, int N) {
  int i = blockIdx.x * blockDim.x + threadIdx.x;
  if (i < N) deg[i] = 1.0f;                       // self-loop weight 1
}

__global__ void k_deg_accum(const int* __restrict__ dst, const float* __restrict__ ew,
                            float* deg, int E) {
  int e = blockIdx.x * blockDim.x + threadIdx.x;
  if (e < E) atomicAdd(&deg[dst[e]], ew[e]);
}

__global__ void k_dinv_selfloop(const float* __restrict__ x, float* degdinv,
                                float* s, int N) {
  int i = blockIdx.x * blockDim.x + threadIdx.x;
  if (i < N) {
    float d  = degdinv[i];
    float di = d > 0.f ? rsqrtf(d) : 0.f;
    degdinv[i] = di;                               // deg -> dinv in place
    s[i] = di * di * x[i];                         // self-loop contribution (ew = 1)
  }
}

__global__ void k_s_accum(const int* __restrict__ src, const int* __restrict__ dst,
                          const float* __restrict__ ew, const float* __restrict__ x,
                          const float* __restrict__ dinv, float* s, int E) {
  int e = blockIdx.x * blockDim.x + threadIdx.x;
  if (e < E) {
    int si = src[e], di = dst[e];
    float w = dinv[si] * ew[e] * dinv[di];
    atomicAdd(&s[di], w * x[si]);
  }
}

// Fold the GRU weights: az = Wz @ Lz[:64], cz = bz @ Lz[:64] + lbz (same for h).
// The R gate and the bottom halves of L* are dead because H0 == 0.
__global__ void k_precompute(const float* __restrict__ Wz, const float* __restrict__ bz,
                             const float* __restrict__ Lz, const float* __restrict__ lbz,
                             const float* __restrict__ Wh, const float* __restrict__ bh,
                             const float* __restrict__ Lh, const float* __restrict__ lbh,
                             float* az, float* cz, float* ah, float* ch) {
  int j = threadIdx.x;                             // 0..63
  float a0 = 0.f, c0 = 0.f, a1 = 0.f, c1 = 0.f;
  for (int k = 0; k < 64; ++k) {
    float lz = Lz[k * 64 + j];
    a0 = fmaf(Wz[k], lz, a0);
    c0 = fmaf(bz[k], lz, c0);
    float lh = Lh[k * 64 + j];
    a1 = fmaf(Wh[k], lh, a1);
    c1 = fmaf(bh[k], lh, c1);
  }
  az[j] = a0; cz[j] = c0 + lbz[j];
  ah[j] = a1; ch[j] = c1 + lbh[j];
}

// ---------------- WMMA node kernel ----------------

__device__ __forceinline__ float fast_sigmoid(float v) {
  return 1.0f / (1.0f + __expf(-v));
}
__device__ __forceinline__ float fast_tanh(float v) {
  float t = __expf(2.0f * v);
  return (t - 1.0f) / (t + 1.0f);
}

__global__ __launch_bounds__(256) void k_tgcn_nodes(
    const float* __restrict__ s,  const float* __restrict__ az, const float* __restrict__ cz,
    const float* __restrict__ ah, const float* __restrict__ ch,
    const float* __restrict__ wout, const float* __restrict__ bout,
    float* __restrict__ out, int N)
{
  const int lane = threadIdx.x & 31;
  const int wave = threadIdx.x >> 5;
  const int base = (blockIdx.x * 8 + wave) * 16;   // 16 nodes (D rows) per wave
  if (base >= N) return;                           // wave-uniform exit: EXEC stays all-ones

  // A (16x4 f32): only K=0 column non-zero, holding per-node scalar s.
  // Layout: VGPR0 lanes 0-15 = (M=lane, K=0); everything else zeroed.
  v2f a; a.x = 0.f; a.y = 0.f;
  if (lane < 16) {
    int node = base + lane;
    if (node < N) a.x = s[node];
  }

  float acc[8];
#pragma unroll
  for (int r = 0; r < 8; ++r) acc[r] = 0.f;

  const int col = lane & 15;                       // D column index N
#pragma unroll
  for (int c = 0; c < 4; ++c) {                    // 4 chunks of 16 hidden dims
    // B (4x16 f32): only K=0 row non-zero = az/ah chunk; other K rows zero,
    // so the result is layout-robust.
    v2f bzv; bzv.x = (lane < 16) ? az[c * 16 + lane] : 0.f; bzv.y = 0.f;
    v2f bhv; bhv.x = (lane < 16) ? ah[c * 16 + lane] : 0.f; bhv.y = 0.f;
    float czv = cz[c * 16 + col];
    float chv = ch[c * 16 + col];
    v8f cc, hh;
#pragma unroll
    for (int r = 0; r < 8; ++r) { cc[r] = czv; hh[r] = chv; }

    // D = s ⊗ a + c  (rank-1 affine pre-activation) on the matrix core, full fp32.
    v8f zp = __builtin_amdgcn_wmma_f32_16x16x4_f32(false, a, false, bzv,
                                                   (short)0, cc, false, false);
    v8f hp = __builtin_amdgcn_wmma_f32_16x16x4_f32(false, a, false, bhv,
                                                   (short)0, hh, false, false);
    float wv = wout[c * 16 + col];
#pragma unroll
    for (int r = 0; r < 8; ++r) {
      float Z  = fast_sigmoid(zp[r]);
      float Ht = fast_tanh(hp[r]);
      float v  = (1.0f - Z) * Ht;                  // Hn, since H0 = 0
      v = fmaxf(v, 0.0f);                          // relu
      acc[r] = fmaf(v, wv, acc[r]);
    }
  }

  // Hidden-dim reduction: sum over the 16 lanes of each half (xor <= 8 stays in-half).
#pragma unroll
  for (int r = 0; r < 8; ++r) {
    float v = acc[r];
    v += __shfl_xor(v, 1, 32);
    v += __shfl_xor(v, 2, 32);
    v += __shfl_xor(v, 4, 32);
    v += __shfl_xor(v, 8, 32);
    acc[r] = v;
  }
  // D layout: lanes 0-15 VGPR r -> row M=r ; lanes 16-31 VGPR r -> row M=8+r.
  if (col == 0) {
    const float bo = bout[0];
    const int mbase = base + ((lane >> 4) ? 8 : 0);
#pragma unroll
    for (int r = 0; r < 8; ++r) {
      int node = mbase + r;
      if (node < N) out[node] = acc[r] + bo;
    }
  }
}

// ---------------- launcher ----------------

extern "C" void kernel_launch(void* const* d_in, const int* in_sizes, int n_in,
                              void* d_out, int out_size, void* d_ws, size_t ws_size,
                              hipStream_t stream) {
  const float* x    = (const float*)d_in[0];
  const float* ew   = (const float*)d_in[1];
  const float* Wz   = (const float*)d_in[2];
  const float* bz   = (const float*)d_in[3];
  const float* Lz   = (const float*)d_in[4];
  const float* lbz  = (const float*)d_in[5];
  // d_in[6..9] (Wr, br, Lr, lbr) are mathematically dead: H0=0 => H*R = 0.
  const float* Wh   = (const float*)d_in[10];
  const float* bh   = (const float*)d_in[11];
  const float* Lh   = (const float*)d_in[12];
  const float* lbh  = (const float*)d_in[13];
  const float* Wout = (const float*)d_in[14];
  const float* bout = (const float*)d_in[15];
  const int*   eidx = (const int*)d_in[16];

  const int N = in_sizes[0];                       // 100000
  const int E = in_sizes[1];                       // 1600000
  const int* src = eidx;                           // edge_index[0]
  const int* dst = eidx + E;                       // edge_index[1]

  float* ws   = (float*)d_ws;
  float* dinv = ws;                                // N floats (deg -> dinv in place)
  float* sagg = ws + (size_t)N;                    // N floats
  float* az   = ws + 2 * (size_t)N;                // 64
  float* cz   = az + 64;                           // 64
  float* ah   = cz + 64;                           // 64
  float* ch   = ah + 64;                           // 64

  const int TB = 256;
  const int gN = (N + TB - 1) / TB;
  const int gE = (E + TB - 1) / TB;

  k_init_deg     <<<gN, TB, 0, stream>>>(dinv, N);
  k_deg_accum    <<<gE, TB, 0, stream>>>(dst, ew, dinv, E);
  k_dinv_selfloop<<<gN, TB, 0, stream>>>(x, dinv, sagg, N);
  k_s_accum      <<<gE, TB, 0, stream>>>(src, dst, ew, x, dinv, sagg, E);
  k_precompute   <<<1, 64, 0, stream>>>(Wz, bz, Lz, lbz, Wh, bh, Lh, lbh, az, cz, ah, ch);

  const int nodesPerBlock = 8 * 16;                // 8 waves x 16 nodes
  const int gMain = (N + nodesPerBlock - 1) / nodesPerBlock;
  k_tgcn_nodes   <<<gMain, TB, 0, stream>>>(sagg, az, cz, ah, ch, Wout, bout,
                                            (float*)d_out, N);
}